// Q3Tensor_67379446940000
// MI455X (gfx1250) — compile-verified
//
#include <hip/hip_runtime.h>

// Q3 unpack: in  = 4096x4096 int32, each holding ten 3-bit fields in bits [29:0]
//            out = 10 planes of 4096x4096, out[k*N + e] = float((in[e] >> (27-3k)) & 7)
//
// Memory-bound: 64 MB read + 671 MB written, write-dominated 10:1.
// Strategy: 1x b128 NT load + 10x b128 NT stores per lane, all coalesced,
// L2 kept clean via non-temporal hints (output is 3.5x the 192MB L2 and
// write-once), plus a global_prefetch_b8 look-ahead on the read stream.

// Native clang vector types (HIP's uint4/float4 are structs and are rejected
// by the nontemporal builtins).
typedef unsigned int v4u __attribute__((ext_vector_type(4)));
typedef float        v4f __attribute__((ext_vector_type(4)));

__global__ __launch_bounds__(256) void q3_unpack_kernel(
    const unsigned int* __restrict__ in, float* __restrict__ out,
    unsigned int n /* total packed elements, multiple of 4 */) {

  const size_t t = (size_t)blockIdx.x * blockDim.x + threadIdx.x;
  const size_t e = t * 4;  // first of 4 consecutive packed int32 handled here
  if (e >= n) return;

  // Prefetch the read stream ~64KB ahead into cache (gfx1250 global_prefetch_b8;
  // speculative, but guard the tail anyway).
  const size_t pf = e + 16384;
  if (pf + 4 <= n) {
    __builtin_prefetch(in + pf, 0, 3);
  }

  // One 16-byte load per lane; read-once -> non-temporal (global_load_b128, TH=NT).
  const v4u p =
      __builtin_nontemporal_load(reinterpret_cast<const v4u*>(in + e));

  const size_t plane = (size_t)n;  // elements per k-plane (step*cols)
  float* o = out + e;

#pragma unroll
  for (int k = 0; k < 10; ++k) {
    const int s = 27 - 3 * k;
    v4f v;
    v.x = (float)((p.x >> s) & 7u);
    v.y = (float)((p.y >> s) & 7u);
    v.z = (float)((p.z >> s) & 7u);
    v.w = (float)((p.w >> s) & 7u);
    // Write-once stream 3.5x larger than L2 -> non-temporal b128 store.
    __builtin_nontemporal_store(v, reinterpret_cast<v4f*>(o + (size_t)k * plane));
  }
}

extern "C" void kernel_launch(void* const* d_in, const int* in_sizes, int n_in,
                              void* d_out, int out_size, void* d_ws, size_t ws_size,
                              hipStream_t stream) {
  (void)n_in; (void)out_size; (void)d_ws; (void)ws_size;

  const unsigned int* packed = (const unsigned int*)d_in[0];
  float* out = (float*)d_out;

  const unsigned int n = (unsigned int)in_sizes[0];  // 4096*4096 = 16,777,216
  const unsigned int n4 = n / 4;                     // int4s = 4,194,304 lanes

  const int block = 256;                              // 8 wave32 per block
  const int grid = (int)((n4 + block - 1) / block);   // 16,384 blocks

  q3_unpack_kernel<<<grid, block, 0, stream>>>(packed, out, n);
}